// MisoFadingChannel_Realvalue_3865470566416
// MI455X (gfx1250) — compile-verified
//
#include <hip/hip_runtime.h>

typedef float v2f __attribute__((ext_vector_type(2)));
typedef float v4f __attribute__((ext_vector_type(4)));
typedef float v8f __attribute__((ext_vector_type(8)));

// ---------------------------------------------------------------------------
// Stage 1: per-batch channel gain via WMMA Gram diagonal.
// Each wave32 handles 16 batches (rows of H, each 8 floats).
// Gram = A * A^T with two V_WMMA_F32_16X16X4_F32 accumulating steps (K=0..3,
// K=4..7). For the symmetric Gram product the 16x4 A layout and 4x16 B layout
// coincide register-for-register, so one v2f load feeds both operands.
// diag(Gram)[m] = ||H_b||^2  ->  g = sqrt(.), inv = 1/g.
// ---------------------------------------------------------------------------
__global__ __launch_bounds__(128) void gain_kernel(const float* __restrict__ H,
                                                   float2* __restrict__ gout,
                                                   int nbatch) {
    const int lane   = threadIdx.x & 31;
    const int waveId = (blockIdx.x * (blockDim.x >> 5)) + (threadIdx.x >> 5);
    const int base   = waveId * 16;
    if (base >= nbatch) return;  // wave-uniform: EXEC stays all-ones for WMMA

    const int m     = lane & 15;                 // row within 16-batch tile
    const int kbase = (lane & 16) ? 2 : 0;       // lanes 16-31 hold K=2,3 (+4)
    const float* h  = H + (size_t)(base + m) * 8 + kbase;

    v2f a0, a1;
    a0[0] = h[0]; a0[1] = h[1];                  // K = kbase, kbase+1
    a1[0] = h[4]; a1[1] = h[5];                  // K = kbase+4, kbase+5

    v8f c = {};
    // 8 args: (neg_a, A, neg_b, B, c_mod, C, reuse_a, reuse_b)
    c = __builtin_amdgcn_wmma_f32_16x16x4_f32(false, a0, false, a0, (short)0, c,
                                              false, false);
    c = __builtin_amdgcn_wmma_f32_16x16x4_f32(false, a1, false, a1, (short)0, c,
                                              false, false);

    // Extract diagonal: element (M,N): lanes<16 -> (v, lane); lanes>=16 -> (v+8, lane-16).
    // diag[M] sits at: lane=M, vgpr=M for M<8; lane=M+16, vgpr=M-8 for M>=8.
    const int v = lane & 7;
    float s2 = c[0];
    s2 = (v == 1) ? c[1] : s2;
    s2 = (v == 2) ? c[2] : s2;
    s2 = (v == 3) ? c[3] : s2;
    s2 = (v == 4) ? c[4] : s2;
    s2 = (v == 5) ? c[5] : s2;
    s2 = (v == 6) ? c[6] : s2;
    s2 = (v == 7) ? c[7] : s2;

    if ((lane < 8) || (lane >= 24)) {
        const float gg = sqrtf(s2);              // g = U*S = ||H_b|| (U = +1)
        gout[base + m] = make_float2(gg, 1.0f / gg);
    }
}

// ---------------------------------------------------------------------------
// Stage 2: Rx = (g*Tx + noise) * inv, 64M elements, pure HBM stream.
// 128-bit non-temporal loads/stores (768 MB streamed once >> 192 MB L2).
// blockIdx.y = batch -> g/inv become wave-uniform scalar loads.
// ---------------------------------------------------------------------------
__global__ __launch_bounds__(256) void rx_kernel(const float* __restrict__ tx,
                                                 const float* __restrict__ nz,
                                                 const float2* __restrict__ gin,
                                                 float* __restrict__ out,
                                                 int quadsPerRow) {
    const int b = blockIdx.y;
    const float2 s = gin[b];
    const float g = s.x, inv = s.y;

    const size_t q = (size_t)b * (size_t)quadsPerRow +
                     (size_t)blockIdx.x * blockDim.x + threadIdx.x;
    const size_t e = q * 4;

    v4f t = __builtin_nontemporal_load((const v4f*)(tx + e));
    v4f n = __builtin_nontemporal_load((const v4f*)(nz + e));
    v4f r;
#pragma unroll
    for (int i = 0; i < 4; ++i) {
        r[i] = fmaf(g, t[i], n[i]) * inv;
    }
    __builtin_nontemporal_store(r, (v4f*)(out + e));
}

extern "C" void kernel_launch(void* const* d_in, const int* in_sizes, int n_in,
                              void* d_out, int out_size, void* d_ws, size_t ws_size,
                              hipStream_t stream) {
    const float* tx = (const float*)d_in[0];   // Tx_sig [B, L]
    const float* H  = (const float*)d_in[1];   // H      [B, 1, 8]
    const float* nz = (const float*)d_in[2];   // noise  [B, L]
    float*  out = (float*)d_out;               // Rx_sig [B, L]
    float2* gws = (float2*)d_ws;               // per-batch {g, 1/g}

    const int B = in_sizes[1] / 8;             // NT = 8
    const int L = (int)((long long)in_sizes[0] / B);

    // Stage 1: one wave32 per 16 batches; 4 waves (128 threads) per block.
    const int waves   = (B + 15) / 16;
    const int blocks1 = (waves + 3) / 4;
    gain_kernel<<<blocks1, 128, 0, stream>>>(H, gws, B);

    // Stage 2: one float4 per thread, one batch row per grid-Y slice.
    const int quads = L / 4;                   // L = 65536 -> 16384 quads/row
    dim3 grid(quads / 256, B);                 // 64 x 1024 blocks of 8 waves
    rx_kernel<<<grid, 256, 0, stream>>>(tx, nz, gws, out, quads);
}